// gaussian_kde_15564961480854
// MI455X (gfx1250) — compile-verified
//
#include <hip/hip_runtime.h>

typedef float v2f __attribute__((ext_vector_type(2)));
typedef float v8f __attribute__((ext_vector_type(8)));

#define KDE_H     0.5f
#define SLICES    8           // waves per block, each owns a slice of the data dim

__device__ __forceinline__ float fast_exp2(float v) {
#if __has_builtin(__builtin_amdgcn_exp2f)
  return __builtin_amdgcn_exp2f(v);
#else
  return exp2f(v);
#endif
}

// One block (8 waves) per 16-query tile. Wave w sweeps data slice w.
// D = A x B with A[16x4] = c*[x^2, x, 1, 0] rows, B[4x16] = [1, -2X, X^2, 0] cols
// gives the 16x16 tile of c*(x_i - X_j)^2 in one V_WMMA_F32_16X16X4_F32,
// where c = -log2(e)/(2h^2), so exp2(D) is the Gaussian kernel directly.
__global__ __launch_bounds__(32 * SLICES)
void kde_pairwise(const float* __restrict__ x, const float* __restrict__ X,
                  float* __restrict__ P, int nd) {
  const int lane = threadIdx.x & 31;
  const int wave = threadIdx.x >> 5;
  const int q0   = blockIdx.x * 16;
  const int m    = lane & 15;          // row (lanes 0-15) / col (all lanes) index

  const float c = -1.4426950408889634f / (2.0f * KDE_H * KDE_H);

  // A fragment (wave-invariant across the whole sweep), with c folded in.
  const float xm = x[q0 + m];
  v2f a;
  a.x = (lane < 16) ? c * xm * xm : c;     // K=0 | K=2
  a.y = (lane < 16) ? c * xm      : 0.0f;  // K=1 | K=3

  v8f acc = {};

  // Uniform trip count (same for every wave) -> scalar loop control.
  const int per_slice = nd / SLICES;
  const int iters     = per_slice / 16;
  const float* Xp     = X + wave * per_slice + m;

  for (int it = 0; it < iters; ++it) {
    const float Xn = Xp[it * 16];
    v2f b;
    b.x = (lane < 16) ? 1.0f       : Xn * Xn;   // K=0 | K=2
    b.y = (lane < 16) ? -2.0f * Xn : 0.0f;      // K=1 | K=3

    v8f zero = {};
    // d[v] = c * squared distance for (row m = v or v+8, col n = lane%16)
    v8f d = __builtin_amdgcn_wmma_f32_16x16x4_f32(
        /*neg_a=*/false, a, /*neg_b=*/false, b,
        /*c_mod=*/(short)0, zero, /*reuse_a=*/false, /*reuse_b=*/false);

#pragma unroll
    for (int v = 0; v < 8; ++v)
      acc[v] += fast_exp2(d[v]);
  }

  // Sum over the data (n) dimension: reduce across each 16-lane half.
#pragma unroll
  for (int v = 0; v < 8; ++v) {
    float s = acc[v];
    s += __shfl_xor(s, 1, 32);
    s += __shfl_xor(s, 2, 32);
    s += __shfl_xor(s, 4, 32);
    s += __shfl_xor(s, 8, 32);
    acc[v] = s;
  }

  __shared__ float lds[SLICES][16];
  if (lane == 0) {
#pragma unroll
    for (int v = 0; v < 8; ++v) lds[wave][v] = acc[v];        // rows 0..7
  } else if (lane == 16) {
#pragma unroll
    for (int v = 0; v < 8; ++v) lds[wave][8 + v] = acc[v];    // rows 8..15
  }
  __syncthreads();

  // Deterministic cross-wave combine + clip.
  if (threadIdx.x < 16) {
    float s = 0.0f;
#pragma unroll
    for (int w = 0; w < SLICES; ++w) s += lds[w][threadIdx.x];
    P[q0 + threadIdx.x] = fmaxf(s, 1e-10f);
  }
}

// Single-block tree reduction: total = sum(P)
__global__ __launch_bounds__(1024)
void kde_sum(const float* __restrict__ P, float* __restrict__ total, int n) {
  __shared__ float lds[32];
  float s = 0.0f;
  for (int i = threadIdx.x; i < n; i += 1024) s += P[i];
#pragma unroll
  for (int msk = 1; msk < 32; msk <<= 1) s += __shfl_xor(s, msk, 32);
  const int lane = threadIdx.x & 31, wave = threadIdx.x >> 5;
  if (lane == 0) lds[wave] = s;
  __syncthreads();
  if (wave == 0) {
    s = lds[lane];
#pragma unroll
    for (int msk = 1; msk < 32; msk <<= 1) s += __shfl_xor(s, msk, 32);
    if (lane == 0) total[0] = s;
  }
}

__global__ void kde_norm(float* __restrict__ P, const float* __restrict__ total, int n) {
  const int i = blockIdx.x * blockDim.x + threadIdx.x;
  if (i < n) P[i] = P[i] / total[0];
}

extern "C" void kernel_launch(void* const* d_in, const int* in_sizes, int n_in,
                              void* d_out, int out_size, void* d_ws, size_t ws_size,
                              hipStream_t stream) {
  (void)n_in; (void)ws_size; (void)out_size;
  const float* x = (const float*)d_in[0];   // queries  [N]
  const float* X = (const float*)d_in[1];   // samples  [M]
  float* out = (float*)d_out;               // P        [N]
  float* tot = (float*)d_ws;                // scalar accumulator

  const int nq = in_sizes[0];               // 16384
  const int nd = in_sizes[1];               // 32768

  kde_pairwise<<<nq / 16, 32 * SLICES, 0, stream>>>(x, X, out, nd);
  kde_sum<<<1, 1024, 0, stream>>>(out, tot, nq);
  kde_norm<<<(nq + 255) / 256, 256, 0, stream>>>(out, tot, nq);
}